// GAT_90520730730897
// MI455X (gfx1250) — compile-verified
//
#include <hip/hip_runtime.h>
#include <hip/hip_bf16.h>

#define Nn   50000
#define Ee   800000
#define INF_ 128
#define NH   128
#define HH   2
#define PP   100000
#define CC   2

typedef float v2f __attribute__((ext_vector_type(2)));
typedef float v4f __attribute__((ext_vector_type(4)));
typedef float v8f __attribute__((ext_vector_type(8)));

static __device__ __forceinline__ void atomicMaxF(float* addr, float val) {
    int* ia = (int*)addr;
    int old = __float_as_int(*addr);
    while (__int_as_float(old) < val) {
        int assumed = old;
        old = atomicCAS(ia, assumed, __float_as_int(val));
        if (old == assumed) break;
    }
}

// ---------------------------------------------------------------------------
// C[M,256] = A[M,128] @ B[128,256], fp32 WMMA 16x16x4. One wave per 16x16 tile.
// A-layout (ISA 7.12.2): lanes 0-15 hold M=0..15 K={k,k+1}; lanes 16-31 K={k+2,k+3}.
// B-layout mirrored; C: VGPR v -> row v (lanes 0-15) / v+8 (lanes 16-31), col = lane&15.
// ---------------------------------------------------------------------------
__global__ __launch_bounds__(256) void gemm_wmma_128x256(
        const float* __restrict__ A, const float* __restrict__ B,
        float* __restrict__ C, int M) {
    int wave = (int)((blockIdx.x * blockDim.x + threadIdx.x) >> 5);
    int lane = threadIdx.x & 31;
    int tm = wave >> 4;            // 256/16 = 16 col tiles
    int tn = wave & 15;
    if (tm * 16 >= M) return;
    int half = lane >> 4;
    int l    = lane & 15;

    const float* Arow = A + (size_t)(tm * 16 + l) * 128;
    const float* Bcol = B + (tn * 16 + l);
    __builtin_prefetch(Bcol, 0, 1);          // global_prefetch_b8 (L2-resident B)

    v8f acc = {0.f, 0.f, 0.f, 0.f, 0.f, 0.f, 0.f, 0.f};
    #pragma unroll 8
    for (int k = 0; k < 128; k += 4) {
        int ka = k + half * 2;
        v2f a = *(const v2f*)(Arow + ka);
        v2f b = { Bcol[(size_t)ka * 256], Bcol[(size_t)(ka + 1) * 256] };
        acc = __builtin_amdgcn_wmma_f32_16x16x4_f32(false, a, false, b,
                                                    (short)0, acc, false, false);
    }
    float* Cp = C + (size_t)(tm * 16 + half * 8) * 256 + tn * 16 + l;
    #pragma unroll
    for (int v = 0; v < 8; ++v) Cp[(size_t)v * 256] = acc[v];
}

// el[n,h] = dot(feat[n,h,:], al[h,:]) ; er likewise  (b128 loads)
__global__ void el_er_k(const float* __restrict__ feat,
                        const float* __restrict__ al, const float* __restrict__ ar,
                        float* __restrict__ el, float* __restrict__ er) {
    int tid = blockIdx.x * blockDim.x + threadIdx.x;
    if (tid >= Nn * HH) return;
    int h = tid & 1;
    const v4f* fp  = (const v4f*)(feat + (size_t)tid * NH);
    const v4f* alp = (const v4f*)(al + h * NH);
    const v4f* arp = (const v4f*)(ar + h * NH);
    float sl = 0.f, sr = 0.f;
    #pragma unroll 4
    for (int f = 0; f < NH / 4; ++f) {
        v4f v = fp[f], va = alp[f], vr = arp[f];
        #pragma unroll
        for (int i = 0; i < 4; ++i) { sl += v[i] * va[i]; sr += v[i] * vr[i]; }
    }
    el[tid] = sl;
    er[tid] = sr;
}

// vectorized fill (n must be a multiple of 4)
__global__ void fill4_k(float* __restrict__ p, long long n4, float v) {
    long long tid = (long long)blockIdx.x * blockDim.x + threadIdx.x;
    if (tid < n4) ((v4f*)p)[tid] = (v4f){v, v, v, v};
}

// e = leaky_relu(el[src]+er[dst]); segment max over dst
__global__ void edge_max_k(const int* __restrict__ src, const int* __restrict__ dst,
                           const float* __restrict__ el, const float* __restrict__ er,
                           float* __restrict__ ebuf, float* __restrict__ emax) {
    int tid = blockIdx.x * blockDim.x + threadIdx.x;
    if (tid >= Ee * HH) return;
    int e = tid >> 1, h = tid & 1;
    int d = dst[e];
    float v = el[src[e] * HH + h] + er[d * HH + h];
    v = (v > 0.f) ? v : 0.2f * v;
    ebuf[tid] = v;
    atomicMaxF(emax + d * HH + h, v);
}

// ee = exp(e - emax[dst]); denom[dst] += ee
__global__ void edge_exp_k(const int* __restrict__ dst,
                           float* __restrict__ ebuf, const float* __restrict__ emax,
                           float* __restrict__ den) {
    int tid = blockIdx.x * blockDim.x + threadIdx.x;
    if (tid >= Ee * HH) return;
    int e = tid >> 1, h = tid & 1;
    int d = dst[e];
    float v = __expf(ebuf[tid] - emax[d * HH + h]);
    ebuf[tid] = v;
    atomicAdd(den + d * HH + h, v);
}

// UNNORMALIZED scatter: agg[dst,h,f] += ee * feat[src,h,f]
// (normalization by den[dst,h]+1e-9 is hoisted to the mean/relu epilogue;
//  relu(x/d) == relu(x)/d for d>0, and den is per-dst constant over the sum)
// One thread per (edge, head, 4 features): b128 feat load + 4 atomic adds.
__global__ void edge_scatter_k(const int* __restrict__ src, const int* __restrict__ dst,
                               const float* __restrict__ feat, const float* __restrict__ ebuf,
                               float* __restrict__ agg) {
    long long tid = (long long)blockIdx.x * blockDim.x + threadIdx.x;
    if (tid >= (long long)Ee * HH * (NH / 4)) return;
    int f4 = (int)(tid & (NH / 4 - 1));
    int eh = (int)(tid >> 5);
    int e = eh >> 1, h = eh & 1;
    int s = src[e], d = dst[e];
    float w = ebuf[eh];
    v4f fv = *(const v4f*)(feat + ((size_t)s * HH + h) * NH + f4 * 4);
    float* ap = agg + ((size_t)d * HH + h) * NH + f4 * 4;
    atomicAdd(ap + 0, w * fv[0]);
    atomicAdd(ap + 1, w * fv[1]);
    atomicAdd(ap + 2, w * fv[2]);
    atomicAdd(ap + 3, w * fv[3]);
}

// hcur[n,f] = 0.5*( relu(agg[n,0,f])/(den[n,0]+1e-9) + relu(agg[n,1,f])/(den[n,1]+1e-9) )
__global__ void mean_relu_k(const float* __restrict__ agg, const float* __restrict__ den,
                            float* __restrict__ hcur) {
    int tid = blockIdx.x * blockDim.x + threadIdx.x;   // over N * NH/4
    if (tid >= Nn * (NH / 4)) return;
    int n = tid >> 5, f4 = tid & 31;
    float r0 = 1.0f / (den[n * HH + 0] + 1e-9f);
    float r1 = 1.0f / (den[n * HH + 1] + 1e-9f);
    v4f a = *(const v4f*)(agg + (size_t)n * (HH * NH) + f4 * 4);
    v4f b = *(const v4f*)(agg + (size_t)n * (HH * NH) + NH + f4 * 4);
    v4f o;
    #pragma unroll
    for (int i = 0; i < 4; ++i)
        o[i] = 0.5f * (fmaxf(a[i], 0.f) * r0 + fmaxf(b[i], 0.f) * r1);
    *(v4f*)(hcur + (size_t)n * NH + f4 * 4) = o;
}

// t[P,128] = relu( z[P,384] @ w1[384,128] + b1 ),   z gathered on the fly
__global__ __launch_bounds__(256) void mlp1_wmma(
        const float* __restrict__ hf, const int* __restrict__ x1, const int* __restrict__ x2,
        const float* __restrict__ w1, const float* __restrict__ b1,
        float* __restrict__ t) {
    int wave = (int)((blockIdx.x * blockDim.x + threadIdx.x) >> 5);
    int lane = threadIdx.x & 31;
    int tm = wave >> 3;            // 128/16 = 8 col tiles
    int tn = wave & 7;
    if (tm * 16 >= PP) return;
    int half = lane >> 4;
    int l    = lane & 15;

    int p = tm * 16 + l;
    const float* h1 = hf + (size_t)x1[p] * NH;
    const float* h2 = hf + (size_t)x2[p] * NH;
    const float* Bcol = w1 + tn * 16 + l;

    v8f acc = {0.f, 0.f, 0.f, 0.f, 0.f, 0.f, 0.f, 0.f};
    for (int k = 0; k < 3 * NH; k += 4) {
        int ka  = k + half * 2;        // segment is wave-uniform: k%4==0, segs 128-aligned
        int seg = ka >> 7, off = ka & 127;
        float a0, a1;
        if (seg == 0)      { a0 = h1[off];                 a1 = h1[off + 1]; }
        else if (seg == 1) { a0 = h2[off];                 a1 = h2[off + 1]; }
        else               { a0 = fabsf(h1[off] - h2[off]);
                             a1 = fabsf(h1[off + 1] - h2[off + 1]); }
        v2f a = { a0, a1 };
        v2f b = { Bcol[(size_t)ka * 128], Bcol[(size_t)(ka + 1) * 128] };
        acc = __builtin_amdgcn_wmma_f32_16x16x4_f32(false, a, false, b,
                                                    (short)0, acc, false, false);
    }
    float bias = b1[tn * 16 + l];
    float* Tp = t + (size_t)(tm * 16 + half * 8) * 128 + tn * 16 + l;
    #pragma unroll
    for (int v = 0; v < 8; ++v) Tp[(size_t)v * 128] = fmaxf(acc[v] + bias, 0.f);
}

// out[p,c] = t[p,:] @ w2[:,c] + b2[c]
__global__ void mlp2_k(const float* __restrict__ t, const float* __restrict__ w2,
                       const float* __restrict__ b2, float* __restrict__ out) {
    int p = blockIdx.x * blockDim.x + threadIdx.x;
    if (p >= PP) return;
    const v4f* tp = (const v4f*)(t + (size_t)p * NH);
    float a0 = b2[0], a1 = b2[1];
    #pragma unroll 2
    for (int k = 0; k < NH / 4; ++k) {
        v4f v  = tp[k];
        v4f w0 = *(const v4f*)(w2 + k * 8);
        v4f w1v = *(const v4f*)(w2 + k * 8 + 4);
        a0 += v[0] * w0[0] + v[1] * w0[2] + v[2] * w1v[0] + v[3] * w1v[2];
        a1 += v[0] * w0[1] + v[1] * w0[3] + v[2] * w1v[1] + v[3] * w1v[3];
    }
    out[p * 2 + 0] = a0;
    out[p * 2 + 1] = a1;
}

static inline int cdiv(long long a, long long b) { return (int)((a + b - 1) / b); }

extern "C" void kernel_launch(void* const* d_in, const int* in_sizes, int n_in,
                              void* d_out, int out_size, void* d_ws, size_t ws_size,
                              hipStream_t stream) {
    const int*   src = (const int*)d_in[0];
    const int*   dst = (const int*)d_in[1];
    const float* h   = (const float*)d_in[2];
    const int*   x1  = (const int*)d_in[3];
    const int*   x2  = (const int*)d_in[4];
    const float* W0  = (const float*)d_in[5];
    const float* al0 = (const float*)d_in[6];
    const float* ar0 = (const float*)d_in[7];
    const float* W1  = (const float*)d_in[8];
    const float* al1 = (const float*)d_in[9];
    const float* ar1 = (const float*)d_in[10];
    const float* w1  = (const float*)d_in[11];
    const float* b1  = (const float*)d_in[12];
    const float* w2  = (const float*)d_in[13];
    const float* b2  = (const float*)d_in[14];
    float* out = (float*)d_out;

    // workspace layout (floats)
    float* ws   = (float*)d_ws;
    float* feat = ws;                                   // N*H*NH = 12.8M
    float* agg  = feat + (size_t)Nn * HH * NH;          // 12.8M
    float* hcur = agg  + (size_t)Nn * HH * NH;          // N*NH = 6.4M
    float* el   = hcur + (size_t)Nn * NH;               // N*H
    float* er   = el   + (size_t)Nn * HH;
    float* emax = er   + (size_t)Nn * HH;
    float* den  = emax + (size_t)Nn * HH;
    float* ebuf = den  + (size_t)Nn * HH;               // E*H = 1.6M
    float* tbuf = feat;                                 // alias: feat dead before MLP

    const int B = 256;
    const int gemm_blocks = cdiv((long long)(Nn / 16) * 16 * 32, B);   // waves*32 threads

    auto layer = [&](const float* hin, const float* W, const float* al, const float* ar) {
        gemm_wmma_128x256<<<gemm_blocks, B, 0, stream>>>(hin, W, feat, Nn);
        el_er_k<<<cdiv(Nn * HH, B), B, 0, stream>>>(feat, al, ar, el, er);
        fill4_k<<<cdiv(Nn * HH / 4, B), B, 0, stream>>>(emax, Nn * HH / 4, -3.0e38f);
        fill4_k<<<cdiv(Nn * HH / 4, B), B, 0, stream>>>(den, Nn * HH / 4, 0.f);
        fill4_k<<<cdiv((long long)Nn * HH * NH / 4, B), B, 0, stream>>>(agg, (long long)Nn * HH * NH / 4, 0.f);
        edge_max_k<<<cdiv(Ee * HH, B), B, 0, stream>>>(src, dst, el, er, ebuf, emax);
        edge_exp_k<<<cdiv(Ee * HH, B), B, 0, stream>>>(dst, ebuf, emax, den);
        edge_scatter_k<<<cdiv((long long)Ee * HH * (NH / 4), B), B, 0, stream>>>(src, dst, feat, ebuf, agg);
        mean_relu_k<<<cdiv(Nn * (NH / 4), B), B, 0, stream>>>(agg, den, hcur);
    };

    layer(h, W0, al0, ar0);
    layer(hcur, W1, al1, ar1);

    const int mlp1_blocks = cdiv((long long)(PP / 16) * 8 * 32, B);
    mlp1_wmma<<<mlp1_blocks, B, 0, stream>>>(hcur, x1, x2, w1, b1, tbuf);
    mlp2_k<<<cdiv(PP, B), B, 0, stream>>>(tbuf, w2, b2, out);
}